// GAT_66924180407033
// MI455X (gfx1250) — compile-verified
//
#include <hip/hip_runtime.h>
#include <hip/hip_bf16.h>
#include <math.h>

typedef float v2f __attribute__((ext_vector_type(2)));
typedef float v8f __attribute__((ext_vector_type(8)));

#define IN_DIM   128
#define HID      32
#define HEADS    2
#define OUT_DIM  32

// ---------------------------------------------------------------------------
// float atomic max via int punning (IEEE ordering trick)
// ---------------------------------------------------------------------------
__device__ __forceinline__ void atomicMaxFloat(float* addr, float value) {
    if (value >= 0.0f) {
        atomicMax((int*)addr, __float_as_int(value));
    } else {
        atomicMin((unsigned int*)addr, __float_as_uint(value));
    }
}

// ---------------------------------------------------------------------------
// fill kernel
// ---------------------------------------------------------------------------
__global__ void fill_kernel(float* __restrict__ p, float v, int n) {
    int i = blockIdx.x * blockDim.x + threadIdx.x;
    if (i < n) p[i] = v;
}

// ---------------------------------------------------------------------------
// WMMA fp32 GEMM:  C[M,NC] = A[M,K] * B[K,NC]
// block = 128 threads (4 waves); each wave computes one 16-row strip across
// all NC columns using v_wmma_f32_16x16x4_f32.
// B is staged in LDS K-pair interleaved:  Bs2[(k/2)*NC + n] = {B[k][n],B[k+1][n]}
// so each lane's B fragment is a single aligned ds_load_b64 straight into an
// even-aligned VGPR pair (no packing moves before the WMMA).
// ---------------------------------------------------------------------------
template <int K, int NC>
__global__ void wmma_gemm_f32(const float* __restrict__ A,
                              const float* __restrict__ B,
                              float* __restrict__ C, int M) {
    constexpr int NT = NC / 16;             // 16-wide column tiles
    __shared__ v2f Bs2[(K / 2) * NC];       // K-pair interleaved B

    // cooperative interleave of B into LDS: pair p = (k2, n)
    for (int p = threadIdx.x; p < (K / 2) * NC; p += blockDim.x) {
        int k2 = p / NC;
        int n  = p - k2 * NC;
        v2f pr;
        pr.x = B[(2 * k2 + 0) * NC + n];
        pr.y = B[(2 * k2 + 1) * NC + n];
        Bs2[p] = pr;
    }
    __syncthreads();

    const int wave = threadIdx.x >> 5;
    const int lane = threadIdx.x & 31;
    const int mt   = blockIdx.x * 4 + wave; // 16-row tile index
    if (mt * 16 >= M) return;               // wave-uniform

    const int halfK = (lane >> 4) * 2;      // 0 for lanes 0-15, 2 for 16-31
    const int ncol  = lane & 15;
    const int arow  = mt * 16 + (lane & 15);
    const float* Arow = A + (size_t)arow * K;

    v8f acc[NT];
#pragma unroll
    for (int t = 0; t < NT; ++t) acc[t] = (v8f){0, 0, 0, 0, 0, 0, 0, 0};

    for (int k = 0; k < K; k += 4) {
        // A fragment: 16x4, lanes 0-15 hold K=k..k+1, lanes 16-31 hold k+2..k+3
        v2f a = *(const v2f*)(Arow + k + halfK);
        const int prow = (k + halfK) >> 1;  // even, so exact pair row
#pragma unroll
        for (int t = 0; t < NT; ++t) {
            v2f b = Bs2[prow * NC + t * 16 + ncol];   // single ds_load_b64
            acc[t] = __builtin_amdgcn_wmma_f32_16x16x4_f32(
                false, a, false, b, (short)0, acc[t], false, false);
        }
    }

    // C/D layout: VGPR j -> row (j + 8*(lane>=16)), col = lane&15
    const int rbase = mt * 16 + ((lane >> 4) << 3);
#pragma unroll
    for (int t = 0; t < NT; ++t) {
#pragma unroll
        for (int j = 0; j < 8; ++j) {
            C[(size_t)(rbase + j) * NC + t * 16 + ncol] = acc[t][j];
        }
    }
}

// ---------------------------------------------------------------------------
// per-node attention scores: asn[n,h] = dot(xl[n,h,:], a_src[h,:]) etc.
// ---------------------------------------------------------------------------
template <int H, int C>
__global__ void score_kernel(const float* __restrict__ xl,
                             const float* __restrict__ a_src,
                             const float* __restrict__ a_dst,
                             float* __restrict__ asn,
                             float* __restrict__ adn, int n) {
    int i = blockIdx.x * blockDim.x + threadIdx.x;
    if (i >= n) return;
    const float* xr = xl + (size_t)i * H * C;
#pragma unroll
    for (int h = 0; h < H; ++h) {
        float s = 0.0f, d = 0.0f;
#pragma unroll
        for (int c = 0; c < C; ++c) {
            float v = xr[h * C + c];
            s += v * a_src[h * C + c];
            d += v * a_dst[h * C + c];
        }
        asn[i * H + h] = s;
        adn[i * H + h] = d;
    }
}

// ---------------------------------------------------------------------------
// edge pass 1: e = leaky_relu(asn[src]+adn[dst], 0.2); segment max into mbuf
// ---------------------------------------------------------------------------
template <int H>
__global__ void edge_max_kernel(const int* __restrict__ ei, int E, int EE,
                                const float* __restrict__ asn,
                                const float* __restrict__ adn,
                                float* __restrict__ ebuf,
                                float* __restrict__ mbuf) {
    int e = blockIdx.x * blockDim.x + threadIdx.x;
    if (e >= EE) return;
    int src, dst;
    if (e < E) { src = ei[e]; dst = ei[E + e]; }
    else       { src = dst = e - E; }
#pragma unroll
    for (int h = 0; h < H; ++h) {
        float v = asn[src * H + h] + adn[dst * H + h];
        v = (v > 0.0f) ? v : 0.2f * v;
        ebuf[(size_t)e * H + h] = v;
        atomicMaxFloat(&mbuf[dst * H + h], v);
    }
}

// ---------------------------------------------------------------------------
// edge pass 2: ex = exp(e - m[dst]); denom[dst] += ex (ebuf overwritten by ex)
// ---------------------------------------------------------------------------
template <int H>
__global__ void edge_exp_kernel(const int* __restrict__ ei, int E, int EE,
                                float* __restrict__ ebuf,
                                const float* __restrict__ mbuf,
                                float* __restrict__ den) {
    int e = blockIdx.x * blockDim.x + threadIdx.x;
    if (e >= EE) return;
    int dst = (e < E) ? ei[E + e] : (e - E);
#pragma unroll
    for (int h = 0; h < H; ++h) {
        float m = mbuf[dst * H + h];
        if (!(m >= -3.4e38f && m <= 3.4e38f)) m = 0.0f;   // isfinite fixup
        float ex = __expf(ebuf[(size_t)e * H + h] - m);
        ebuf[(size_t)e * H + h] = ex;
        atomicAdd(&den[dst * H + h], ex);
    }
}

// ---------------------------------------------------------------------------
// edge pass 3: agg[dst] += xl[src] * alpha   (one thread = one edge x float4)
// ---------------------------------------------------------------------------
template <int H, int C>
__global__ void edge_agg_kernel(const int* __restrict__ ei, int E, int EE,
                                const float* __restrict__ ebuf,
                                const float* __restrict__ den,
                                const float* __restrict__ xl,
                                float* __restrict__ agg) {
    constexpr int F = H * C;
    constexpr int G = F / 4;     // float4 chunks per edge
    long long t = (long long)blockIdx.x * blockDim.x + threadIdx.x;
    long long e = t / G;
    int g = (int)(t % G);
    if (e >= EE) return;
    int src, dst;
    if (e < E) { src = ei[e]; dst = ei[E + e]; }
    else       { src = dst = (int)(e - E); }
    int f0 = g * 4;
    int h  = f0 / C;             // constant over the 4-float chunk (C%4==0)
    float alpha = ebuf[(size_t)e * H + h] / (den[dst * H + h] + 1e-16f);
    float4 v = *(const float4*)(xl + (size_t)src * F + f0);
    float* o = agg + (size_t)dst * F + f0;
    atomicAdd(o + 0, v.x * alpha);
    atomicAdd(o + 1, v.y * alpha);
    atomicAdd(o + 2, v.z * alpha);
    atomicAdd(o + 3, v.w * alpha);
}

// ---------------------------------------------------------------------------
// elementwise: a = relu(a + b[f])  /  a = a + b[f]
// ---------------------------------------------------------------------------
__global__ void bias_relu_kernel(float* __restrict__ a,
                                 const float* __restrict__ b, int n, int F) {
    int i = blockIdx.x * blockDim.x + threadIdx.x;
    if (i >= n * F) return;
    float v = a[i] + b[i % F];
    a[i] = v > 0.0f ? v : 0.0f;
}

__global__ void bias_add_kernel(float* __restrict__ a,
                                const float* __restrict__ b, int n, int F) {
    int i = blockIdx.x * blockDim.x + threadIdx.x;
    if (i >= n * F) return;
    a[i] += b[i % F];
}

// ---------------------------------------------------------------------------
// launch
// ---------------------------------------------------------------------------
static inline int cdiv(long long a, long long b) { return (int)((a + b - 1) / b); }

extern "C" void kernel_launch(void* const* d_in, const int* in_sizes, int n_in,
                              void* d_out, int out_size, void* d_ws, size_t ws_size,
                              hipStream_t stream) {
    const float* x      = (const float*)d_in[0];
    const int*   ei     = (const int*)d_in[1];
    const float* W1     = (const float*)d_in[2];
    const float* a_src1 = (const float*)d_in[3];
    const float* a_dst1 = (const float*)d_in[4];
    const float* b1     = (const float*)d_in[5];
    const float* W2     = (const float*)d_in[6];
    const float* a_src2 = (const float*)d_in[7];
    const float* a_dst2 = (const float*)d_in[8];
    const float* b2     = (const float*)d_in[9];
    float* out = (float*)d_out;

    const int N  = in_sizes[0] / IN_DIM;          // 100000
    const int E  = in_sizes[1] / 2;               // 1600000
    const int EE = E + N;                         // + self loops
    const int F1 = HEADS * HID;                   // 64

    // workspace layout
    float* ws   = (float*)d_ws;
    float* xl1  = ws;                         // N*64
    float* agg1 = xl1  + (size_t)N * F1;      // N*64 (becomes h after bias+relu)
    float* asn1 = agg1 + (size_t)N * F1;      // N*2
    float* adn1 = asn1 + (size_t)N * HEADS;   // N*2
    float* m1   = adn1 + (size_t)N * HEADS;   // N*2
    float* den1 = m1   + (size_t)N * HEADS;   // N*2
    float* ex1  = den1 + (size_t)N * HEADS;   // EE*2
    float* xl2  = ex1  + (size_t)EE * HEADS;  // N*32
    float* asn2 = xl2  + (size_t)N * OUT_DIM; // N
    float* adn2 = asn2 + (size_t)N;           // N
    float* m2   = adn2 + (size_t)N;           // N
    float* den2 = m2   + (size_t)N;           // N
    float* ex2  = den2 + (size_t)N;           // EE

    const int TB = 256;

    // ---- init accumulators ----
    fill_kernel<<<cdiv((long long)N * F1, TB), TB, 0, stream>>>(agg1, 0.0f, N * F1);
    fill_kernel<<<cdiv((long long)N * OUT_DIM, TB), TB, 0, stream>>>(out, 0.0f, N * OUT_DIM);
    fill_kernel<<<cdiv((long long)N * HEADS, TB), TB, 0, stream>>>(m1, -INFINITY, N * HEADS);
    fill_kernel<<<cdiv((long long)N * HEADS, TB), TB, 0, stream>>>(den1, 0.0f, N * HEADS);
    fill_kernel<<<cdiv(N, TB), TB, 0, stream>>>(m2, -INFINITY, N);
    fill_kernel<<<cdiv(N, TB), TB, 0, stream>>>(den2, 0.0f, N);

    // ---- layer 1 ----
    {
        int mtiles = cdiv(N, 16);
        wmma_gemm_f32<IN_DIM, F1><<<cdiv(mtiles, 4), 128, 0, stream>>>(x, W1, xl1, N);
        score_kernel<HEADS, HID><<<cdiv(N, TB), TB, 0, stream>>>(xl1, a_src1, a_dst1, asn1, adn1, N);
        edge_max_kernel<HEADS><<<cdiv(EE, TB), TB, 0, stream>>>(ei, E, EE, asn1, adn1, ex1, m1);
        edge_exp_kernel<HEADS><<<cdiv(EE, TB), TB, 0, stream>>>(ei, E, EE, ex1, m1, den1);
        edge_agg_kernel<HEADS, HID><<<cdiv((long long)EE * (F1 / 4), TB), TB, 0, stream>>>(
            ei, E, EE, ex1, den1, xl1, agg1);
        bias_relu_kernel<<<cdiv((long long)N * F1, TB), TB, 0, stream>>>(agg1, b1, N, F1);
    }

    // ---- layer 2 ----
    {
        int mtiles = cdiv(N, 16);
        wmma_gemm_f32<F1, OUT_DIM><<<cdiv(mtiles, 4), 128, 0, stream>>>(agg1, W2, xl2, N);
        score_kernel<1, OUT_DIM><<<cdiv(N, TB), TB, 0, stream>>>(xl2, a_src2, a_dst2, asn2, adn2, N);
        edge_max_kernel<1><<<cdiv(EE, TB), TB, 0, stream>>>(ei, E, EE, asn2, adn2, ex2, m2);
        edge_exp_kernel<1><<<cdiv(EE, TB), TB, 0, stream>>>(ei, E, EE, ex2, m2, den2);
        edge_agg_kernel<1, OUT_DIM><<<cdiv((long long)EE * (OUT_DIM / 4), TB), TB, 0, stream>>>(
            ei, E, EE, ex2, den2, xl2, out);
        bias_add_kernel<<<cdiv((long long)N * OUT_DIM, TB), TB, 0, stream>>>(out, b2, N, OUT_DIM);
    }
}